// GridSampler_57449482551853
// MI455X (gfx1250) — compile-verified
//
#include <hip/hip_runtime.h>

// Bilinear grid sampler for MI455X (gfx1250).
// im:   [B,H,W,C] NHWC f32,  B=8 H=512 W=512 C=16
// grid: [B,2,H,W] f32, coords in [-1,1]
// out:  [B,H,W,C] f32
//
// Memory-bound gather kernel (~272 MB min HBM traffic -> ~12us @ 23.3 TB/s).
//  * 4 lanes per pixel, one float4 (b128) each -> every 64B corner row is one
//    contiguous coalesced segment per pixel.
//  * Block = 256 threads = 64 consecutive pixels of one row; the block's
//    coordinate tile (2 x 256B contiguous) is fetched with a single
//    global_load_async_to_lds_b128 (wave 0, 32 lanes x 16B = 512B), tracked
//    by ASYNCcnt -- removes the 4x-redundant per-lane coordinate loads.
//  * im gathers stay temporal (16 MB/image << 192 MB L2, ~4x reuse per texel);
//    output stores are non-temporal (streamed once).

typedef __attribute__((ext_vector_type(4))) float v4f;

constexpr int B = 8;
constexpr int H = 512;
constexpr int W = 512;
constexpr int C = 16;
constexpr int HW = H * W;                              // 262144
constexpr unsigned NPIX = (unsigned)B * HW;            // 2,097,152
constexpr unsigned PIX_PER_BLOCK = 64;                 // 256 threads / 4 lanes
constexpr unsigned NBLOCKS = NPIX / PIX_PER_BLOCK;     // 32768

__global__ __launch_bounds__(256) void GridSampler_57449482551853_kernel(
    const float* __restrict__ im,
    const float* __restrict__ grid,
    float* __restrict__ out)
{
    __shared__ float sgrid[2 * PIX_PER_BLOCK];   // 512 B: [0..63]=gx, [64..127]=gy

    const unsigned tb   = threadIdx.x;
    const unsigned pixb = blockIdx.x * PIX_PER_BLOCK;   // first pixel of block

    // ---- wave 0: async-copy this block's coordinate tile into LDS ----
    if (tb < 32u) {
        unsigned lane = tb;
        unsigned bb = pixb >> 18;
        unsigned yy = (pixb >> 9) & (H - 1);
        unsigned xs = pixb & (W - 1);                    // multiple of 64
        unsigned gidx0 = bb * (2u * HW) + yy * W + xs;   // x-plane tile start (elems)
        // lanes 0-15 -> x-plane (256B), lanes 16-31 -> y-plane (256B)
        unsigned elem = (lane < 16u) ? (gidx0 + lane * 4u)
                                     : (gidx0 + (unsigned)HW + (lane - 16u) * 4u);
        unsigned voff   = elem * 4u;                     // byte offset from grid base
        // generic shared ptr: bits[31:0] == workgroup-relative LDS byte offset
        unsigned ldsoff = (unsigned)(uintptr_t)&sgrid[0] + lane * 16u;
        asm volatile("global_load_async_to_lds_b128 %0, %1, %2"
                     :
                     : "v"(ldsoff), "v"(voff), "s"(grid)
                     : "memory");
        asm volatile("s_wait_asynccnt 0x0" ::: "memory");
    }
    __syncthreads();

    const unsigned p = tb >> 2;      // pixel within block
    const unsigned q = tb & 3u;      // channel quarter: channels [4q, 4q+4)
    const unsigned pix = pixb + p;

    float gxn = sgrid[p];                      // ds_load_b32 (broadcast across 4 lanes)
    float gyn = sgrid[PIX_PER_BLOCK + p];

    // map [-1,1] -> pixel space
    float gx = (gxn + 1.0f) * (0.5f * (float)W);
    float gy = (gyn + 1.0f) * (0.5f * (float)H);

    float x0f = floorf(gx);
    float y0f = floorf(gy);
    float x1f = x0f + 1.0f;
    float y1f = y0f + 1.0f;

    // clip exactly like the reference (v_med3_num_f32)
    x0f = fminf(fmaxf(x0f, 0.0f), (float)(W - 2));
    x1f = fminf(fmaxf(x1f, 1.0f), (float)(W - 1));
    y0f = fminf(fmaxf(y0f, 0.0f), (float)(H - 2));
    y1f = fminf(fmaxf(y1f, 1.0f), (float)(H - 1));

    int x0 = (int)x0f, x1 = (int)x1f, y0 = (int)y0f, y1 = (int)y1f;

    // bilinear weights; (x1-x0)*(y1-y0) == 1 after clipping -> skip divide
    float wx0 = x1f - gx;
    float wx1 = gx  - x0f;
    float wy0 = y1f - gy;
    float wy1 = gy  - y0f;
    float w00 = wx0 * wy0;
    float w01 = wx0 * wy1;
    float w10 = wx1 * wy0;
    float w11 = wx1 * wy1;

    // corner element offsets: ((b*H + yc)*W + xc)*C + 4*q   (16B aligned)
    unsigned b   = pix >> 18;
    unsigned bhw = b * (unsigned)HW;
    unsigned ch  = q * 4u;
    unsigned row0 = bhw + (unsigned)y0 * W;
    unsigned row1 = bhw + (unsigned)y1 * W;
    unsigned o00 = ((row0 + (unsigned)x0) << 4) + ch;
    unsigned o10 = ((row0 + (unsigned)x1) << 4) + ch;
    unsigned o01 = ((row1 + (unsigned)x0) << 4) + ch;
    unsigned o11 = ((row1 + (unsigned)x1) << 4) + ch;

    // 4x global_load_b128 gathers (temporal: corner reuse lands in L2)
    v4f Q00 = *(const v4f*)(im + o00);
    v4f Q10 = *(const v4f*)(im + o10);
    v4f Q01 = *(const v4f*)(im + o01);
    v4f Q11 = *(const v4f*)(im + o11);

    v4f r;
    r.x = fmaf(w00, Q00.x, fmaf(w01, Q01.x, fmaf(w10, Q10.x, w11 * Q11.x)));
    r.y = fmaf(w00, Q00.y, fmaf(w01, Q01.y, fmaf(w10, Q10.y, w11 * Q11.y)));
    r.z = fmaf(w00, Q00.z, fmaf(w01, Q01.z, fmaf(w10, Q10.z, w11 * Q11.z)));
    r.w = fmaf(w00, Q00.w, fmaf(w01, Q01.w, fmaf(w10, Q10.w, w11 * Q11.w)));

    // streaming store, non-temporal
    unsigned oo = (pix << 4) + ch;
    __builtin_nontemporal_store(r, (v4f*)(out + oo));
}

extern "C" void kernel_launch(void* const* d_in, const int* in_sizes, int n_in,
                              void* d_out, int out_size, void* d_ws, size_t ws_size,
                              hipStream_t stream) {
    (void)in_sizes; (void)n_in; (void)d_ws; (void)ws_size; (void)out_size;
    const float* im   = (const float*)d_in[0];
    const float* grid = (const float*)d_in[1];
    float* out = (float*)d_out;

    GridSampler_57449482551853_kernel<<<NBLOCKS, 256, 0, stream>>>(im, grid, out);
}